// MCGRMRetrieval_31215822307418
// MI455X (gfx1250) — compile-verified
//
#include <hip/hip_runtime.h>
#include <hip/hip_bf16.h>
#include <math.h>

// Problem constants (match reference)
#define BB 4
#define TT 2048
#define NN 32
#define DD 1024
#define HH 16
#define DKK 64
#define DVV 64
#define SEG 64

typedef float v2f __attribute__((ext_vector_type(2)));
typedef float v8f __attribute__((ext_vector_type(8)));

#define WMMA_F32(A, Bv, C) \
    __builtin_amdgcn_wmma_f32_16x16x4_f32(false, (A), false, (Bv), (short)0, (C), false, false)

__device__ __forceinline__ float wave_sum(float v) {
#pragma unroll
    for (int off = 16; off > 0; off >>= 1) v += __shfl_xor(v, off, 32);
    return v;
}
__device__ __forceinline__ float wave_max(float v) {
#pragma unroll
    for (int off = 16; off > 0; off >>= 1) v = fmaxf(v, __shfl_xor(v, off, 32));
    return v;
}

// ---------------------------------------------------------------------------
// Kernel 1: u = x @ W_u^T   (8192 x 1024) = (8192 x 1024) * (1024 x 1024)^T
// grid (64, 16), block 256 (8 waves). Wave w -> 16 rows, full 64-col strip.
// W tile kept row-major in LDS: the (k0,k0+1) B-fragment pair is contiguous,
// so each fragment is one aligned ds_load_b64 (no VGPR repacking).
// ---------------------------------------------------------------------------
__global__ __launch_bounds__(256) void ugemm_kernel(const float* __restrict__ x,
                                                    const float* __restrict__ W,
                                                    float* __restrict__ u) {
    __shared__ __align__(16) float xs[128 * 68];  // x tile, row pitch 68
    __shared__ __align__(16) float wt[64 * 68];   // W tile row-major: wt[j][k]

    const int tid  = threadIdx.x;
    const int rb   = blockIdx.x * 128;   // row block
    const int jb   = blockIdx.y * 64;    // col block
    const int wave = tid >> 5, lane = tid & 31;
    const int m    = lane & 15, half = lane >> 4;
    const int r0   = wave * 16;

    v8f acc0 = {}, acc1 = {}, acc2 = {}, acc3 = {};

    for (int kc = 0; kc < DD; kc += 64) {
        // stage x tile: 128x64 floats, 8 float4 per thread (coalesced both ways)
#pragma unroll
        for (int i = 0; i < 8; ++i) {
            int idx = tid + i * 256;           // 0..2047
            int r = idx >> 4, c4 = idx & 15;
            float4 v = *(const float4*)(x + (size_t)(rb + r) * DD + kc + c4 * 4);
            *(float4*)(xs + r * 68 + c4 * 4) = v;
        }
        // stage W tile row-major: wt[j][k] = W[jb+j][kc+k]
#pragma unroll
        for (int i = 0; i < 4; ++i) {
            int idx = tid + i * 256;           // 0..1023
            int j = idx >> 4, k4 = idx & 15;
            float4 v = *(const float4*)(W + (size_t)(jb + j) * DD + kc + k4 * 4);
            *(float4*)(wt + j * 68 + k4 * 4) = v;
        }
        __syncthreads();

        const float* xrow = xs + (r0 + m) * 68;
        const float* wrow = wt + m * 68;       // row j = v*16 + m, tile offset added below
#pragma unroll
        for (int s = 0; s < 16; ++s) {
            const int k0 = 4 * s + 2 * half;
            v2f a  = *(const v2f*)(xrow + k0);
            v2f b0 = *(const v2f*)(wrow + k0);
            v2f b1 = *(const v2f*)(wrow + 16 * 68 + k0);
            v2f b2 = *(const v2f*)(wrow + 32 * 68 + k0);
            v2f b3 = *(const v2f*)(wrow + 48 * 68 + k0);
            acc0 = WMMA_F32(a, b0, acc0);
            acc1 = WMMA_F32(a, b1, acc1);
            acc2 = WMMA_F32(a, b2, acc2);
            acc3 = WMMA_F32(a, b3, acc3);
        }
        __syncthreads();
    }
    // C layout: VGPR i -> row M=i (lanes 0-15) / M=i+8 (lanes 16-31), N = m
#pragma unroll
    for (int i = 0; i < 8; ++i) {
        int r = rb + r0 + i + 8 * half;
        float* urow = u + (size_t)r * DD + jb + m;
        urow[0]  = acc0[i];
        urow[16] = acc1[i];
        urow[32] = acc2[i];
        urow[48] = acc3[i];
    }
}

// ---------------------------------------------------------------------------
// Kernel 2: scores + masked softmax. One wave per token; 8 tokens/block.
// Writes attn_weights (33/token) and gate_current straight into d_out.
// ---------------------------------------------------------------------------
__global__ __launch_bounds__(256) void softmax_kernel(const float* __restrict__ u,
                                                      const float* __restrict__ m_stack,
                                                      const float* __restrict__ m_current,
                                                      float* __restrict__ gate_cur,
                                                      float* __restrict__ attn) {
    const int wave = threadIdx.x >> 5, lane = threadIdx.x & 31;
    const int g = blockIdx.x * 8 + wave;      // token id in [0, B*T)
    const int b = g >> 11, t = g & (TT - 1);
    const float scale = 1.0f / 32.0f;         // 1/sqrt(1024)

    float ureg[32];
    const float* urow = u + (size_t)g * DD;
#pragma unroll
    for (int i = 0; i < 32; ++i) ureg[i] = urow[lane + 32 * i];

    float sval = -INFINITY;                   // this lane's score for n == lane
    for (int n = 0; n < NN; ++n) {
        const float* mrow = m_stack + (size_t)(b * NN + n) * DD;
        float acc = 0.f;
#pragma unroll
        for (int i = 0; i < 32; ++i) acc = fmaf(ureg[i], mrow[lane + 32 * i], acc);
        acc = wave_sum(acc);
        if (lane == n) sval = (t >= (n + 1) * SEG) ? acc * scale : -INFINITY;
    }
    // current-score dot
    float accc = 0.f;
    const float* crow = m_current + (size_t)g * DD;
#pragma unroll
    for (int i = 0; i < 32; ++i) accc = fmaf(ureg[i], crow[lane + 32 * i], accc);
    const float scur = wave_sum(accc) * scale;   // finite, broadcast to all lanes

    float mx = fmaxf(wave_max(sval), scur);      // mx is finite (scur finite)
    float e  = expf(sval - mx);                  // 0 for masked lanes
    float tot = wave_sum(e);
    float ecur = expf(scur - mx);
    float inv = 1.0f / (tot + ecur);

    attn[(size_t)g * (NN + 1) + lane] = e * inv;
    if (lane == 0) {
        attn[(size_t)g * (NN + 1) + NN] = ecur * inv;
        gate_cur[g] = ecur * inv;
    }
}

// ---------------------------------------------------------------------------
// Kernel 3: h[b,t,h,v] = sum_n gate[b,t,n] * (phi[b,t,h,:] @ S[b,n,h,:,:])
// Block = (t-tile of 128, h, b); 8 waves, each a 16x64 output strip.
// State tile transposed in LDS (stt[v][k]) so (k0,k0+1) B-pairs are contiguous
// -> one ds_load_b64 per fragment. Gate folded into the A fragment per n.
// ---------------------------------------------------------------------------
__global__ __launch_bounds__(256) void retrieve_kernel(const float* __restrict__ phi,
                                                       const float* __restrict__ state,
                                                       const float* __restrict__ attn,
                                                       float* __restrict__ hout) {
    __shared__ __align__(16) float ph[128 * 68];  // phi tile, 128 t-rows x 64 k
    __shared__ __align__(16) float stt[64 * 68];  // state tile transposed: stt[v][k]

    const int tb = blockIdx.x * 128;
    const int h  = blockIdx.y;
    const int b  = blockIdx.z;
    const int tid = threadIdx.x, wave = tid >> 5, lane = tid & 31;
    const int m = lane & 15, half = lane >> 4, r0 = wave * 16;

    // stage phi tile
#pragma unroll
    for (int i = 0; i < 8; ++i) {
        int idx = tid + i * 256;
        int r = idx >> 4, c4 = idx & 15;
        float4 v = *(const float4*)(phi +
            ((size_t)((b * TT + tb + r) * HH + h)) * DKK + c4 * 4);
        *(float4*)(ph + r * 68 + c4 * 4) = v;
    }
    __syncthreads();

    // unscaled A fragments for this wave's 16 rows (full K=64)
    v2f af[16];
    {
        const float* xrow = ph + (r0 + m) * 68;
#pragma unroll
        for (int s = 0; s < 16; ++s) {
            int k0 = 4 * s + 2 * half;
            af[s] = *(const v2f*)(xrow + k0);
        }
    }

    v8f acc0 = {}, acc1 = {}, acc2 = {}, acc3 = {};
    const int ncnt = tb / SEG + 1;   // segments that can have non-zero gate here

    for (int n = 0; n < ncnt; ++n) {
        __syncthreads();   // previous stt reads done
        // stage state tile S[b,n,h] transposed: stt[v][k] = S[k][v]
#pragma unroll
        for (int i = 0; i < 4; ++i) {
            int idx = tid + i * 256;
            int k = idx >> 4, c4 = idx & 15;
            float4 v = *(const float4*)(state +
                (((size_t)(b * NN + n) * HH + h) * DKK + k) * DVV + c4 * 4);
            stt[(c4 * 4 + 0) * 68 + k] = v.x;
            stt[(c4 * 4 + 1) * 68 + k] = v.y;
            stt[(c4 * 4 + 2) * 68 + k] = v.z;
            stt[(c4 * 4 + 3) * 68 + k] = v.w;
        }
        __syncthreads();

        const float g = attn[(size_t)(b * TT + tb + r0 + m) * (NN + 1) + n];
        const float* srow = stt + m * 68;   // row v = vtile*16 + m
#pragma unroll
        for (int s = 0; s < 16; ++s) {
            const int k0 = 4 * s + 2 * half;
            v2f a = af[s];
            a.x *= g; a.y *= g;
            v2f b0 = *(const v2f*)(srow + k0);
            v2f b1 = *(const v2f*)(srow + 16 * 68 + k0);
            v2f b2 = *(const v2f*)(srow + 32 * 68 + k0);
            v2f b3 = *(const v2f*)(srow + 48 * 68 + k0);
            acc0 = WMMA_F32(a, b0, acc0);
            acc1 = WMMA_F32(a, b1, acc1);
            acc2 = WMMA_F32(a, b2, acc2);
            acc3 = WMMA_F32(a, b3, acc3);
        }
    }

#pragma unroll
    for (int i = 0; i < 8; ++i) {
        int t = tb + r0 + i + 8 * half;
        float* orow = hout + ((size_t)(b * TT + t)) * (HH * DVV) + h * DVV + m;
        orow[0]  = acc0[i];
        orow[16] = acc1[i];
        orow[32] = acc2[i];
        orow[48] = acc3[i];
    }
}

// ---------------------------------------------------------------------------
extern "C" void kernel_launch(void* const* d_in, const int* in_sizes, int n_in,
                              void* d_out, int out_size, void* d_ws, size_t ws_size,
                              hipStream_t stream) {
    const float* x           = (const float*)d_in[0];
    const float* phi_q       = (const float*)d_in[1];
    const float* state_stack = (const float*)d_in[2];
    const float* m_stack     = (const float*)d_in[3];
    const float* m_current   = (const float*)d_in[4];
    // d_in[5] = causal_mask: recomputed analytically in-kernel (t >= (n+1)*SEG)
    const float* W_u         = (const float*)d_in[6];

    float* out      = (float*)d_out;
    float* h_ret    = out;                                     // B*T*H*DV  = 8388608
    float* gate_cur = out + (size_t)BB * TT * HH * DVV;        // B*T       = 8192
    float* attn     = gate_cur + (size_t)BB * TT;              // B*T*(N+1) = 270336

    float* u = (float*)d_ws;                                   // B*T*D floats = 32 MB

    dim3 gA(64, 16);
    ugemm_kernel<<<gA, 256, 0, stream>>>(x, W_u, u);

    softmax_kernel<<<1024, 256, 0, stream>>>(u, m_stack, m_current, gate_cur, attn);

    dim3 gC(TT / 128, HH, BB);
    retrieve_kernel<<<gC, 256, 0, stream>>>(phi_q, state_stack, attn, h_ret);
}